// Clahe_76982993814155
// MI455X (gfx1250) — compile-verified
//
#include <hip/hip_runtime.h>
#include <hip/hip_bf16.h>

// ---------------------------------------------------------------------------
// CLAHE (OpenCV-style, 8x8 tiles, 256 bins) on Lab-L of a 2048x2048 RGB image.
//
// MI455X reasoning: workload is bandwidth-bound (~144 MB min traffic @ 23.3
// TB/s ~ 6us; only ~0.6 GFLOP total). Pixel passes are streaming VALU with
// coalesced B32 loads/stores; Lab is recomputed in the apply pass to avoid
// 72 MB of intermediate traffic. The per-tile 256-bin cumsum is expressed as
// a 16x16 matmul against an upper-triangular ones matrix and runs on
// v_wmma_f32_16x16x4_f32 (one wave32 per tile, K-loop of 4) -- exact in f32
// for counts <= 65536.
// ---------------------------------------------------------------------------

#define IMG_H 2048
#define IMG_W 2048
#define NT    8
#define TILE  256           // tile height == width
#define NBINS 256
#define AREA_F 65536.0f     // 256*256
#define CLIP_V 2560.0f      // max(10 * 65536 / 256, 1)
#define LAB_EPS 0.008856f

typedef __attribute__((ext_vector_type(2))) float v2f;
typedef __attribute__((ext_vector_type(8))) float v8f;

// ----------------------------- color math ---------------------------------
__device__ __forceinline__ void rgb2lab_f(float r, float g, float b,
                                          float& L, float& A, float& B) {
    // xyz = M_RGB2XYZ * rgb / whitepoint
    float x = (0.412453f * r + 0.357580f * g + 0.180423f * b) * (1.0f / 0.950456f);
    float y = (0.212671f * r + 0.715160f * g + 0.072169f * b);
    float z = (0.019334f * r + 0.119193f * g + 0.950227f * b) * (1.0f / 1.088754f);
    float fx = (x > LAB_EPS) ? cbrtf(x) : 7.787f * x + (16.0f / 116.0f);
    float fy = (y > LAB_EPS) ? cbrtf(y) : 7.787f * y + (16.0f / 116.0f);
    float fz = (z > LAB_EPS) ? cbrtf(z) : 7.787f * z + (16.0f / 116.0f);
    L = 116.0f * fy - 16.0f;
    A = 500.0f * (fx - fy);
    B = 200.0f * (fy - fz);
}

__device__ __forceinline__ int lab_L_to_bin(float L) {
    float v = fminf(fmaxf(roundf(L * (255.0f / 100.0f)), 0.0f), 255.0f);
    return (int)v;
}

// ------------------------------ kernel 0: zero ws --------------------------
__global__ void k_zero(unsigned int* __restrict__ p, int n) {
    int i = blockIdx.x * blockDim.x + threadIdx.x;
    if (i < n) p[i] = 0u;
}

// --------------------- kernel 1: per-tile histograms -----------------------
// 1024 blocks * 256 threads. Block b handles tile (b>>4), row-segment (b&15):
// 16 rows x 256 cols. LDS histogram, flushed via global integer atomics.
__global__ void k_hist(const float* __restrict__ x, unsigned int* __restrict__ ghist) {
    __shared__ unsigned int lh[NBINS];
    const int tid = threadIdx.x;
    lh[tid] = 0u;
    __syncthreads();

    const int bid  = blockIdx.x;
    const int tile = bid >> 4;       // 0..63
    const int seg  = bid & 15;       // 0..15
    const int ty   = tile >> 3;
    const int tx   = tile & 7;
    const int col  = tx * TILE + tid;
    const int row0 = ty * TILE + seg * 16;

    const float* __restrict__ R = x;
    const float* __restrict__ G = x + IMG_H * IMG_W;
    const float* __restrict__ B = x + 2 * IMG_H * IMG_W;

    #pragma unroll 4
    for (int r = 0; r < 16; ++r) {
        int idx = (row0 + r) * IMG_W + col;
        float L, a, b;
        rgb2lab_f(R[idx], G[idx], B[idx], L, a, b);
        atomicAdd(&lh[lab_L_to_bin(L)], 1u);
    }
    __syncthreads();
    atomicAdd(&ghist[tile * NBINS + tid], lh[tid]);
}

// ------------- kernel 2: clip + redistribute + WMMA cumsum -> LUT ----------
// One wave32 per tile (64 blocks x 32 threads).
// Histogram viewed as 16x16 matrix H (row r = bins 16r..16r+15):
//   cdf = H x U  (U = upper-triangular ones, inclusive in-row prefix)
//       + offArr[row] (exclusive prefix of row sums, via shfl scan)
// H x U is 4 accumulating v_wmma_f32_16x16x4_f32 steps.
__global__ void k_lut(const unsigned int* __restrict__ ghist, float* __restrict__ glut) {
    __shared__ float hh[NBINS];
    __shared__ float offArr[16];
    const int l    = threadIdx.x;    // 0..31
    const int tile = blockIdx.x;     // 0..63

    // clip + uniform redistribution of excess
    float raw[8];
    float ex = 0.0f;
    #pragma unroll
    for (int i = 0; i < 8; ++i) {
        float v = (float)ghist[tile * NBINS + l * 8 + i];
        raw[i] = v;
        ex += fmaxf(v - CLIP_V, 0.0f);
    }
    #pragma unroll
    for (int m = 16; m >= 1; m >>= 1) ex += __shfl_xor(ex, m, 32);
    const float bonus = ex * (1.0f / (float)NBINS);
    #pragma unroll
    for (int i = 0; i < 8; ++i)
        hh[l * 8 + i] = fminf(raw[i], CLIP_V) + bonus;
    __syncthreads();

    // exclusive prefix of row sums over 16 rows (lanes 0..15)
    if (l < 16) {
        float rs = 0.0f;
        #pragma unroll
        for (int c2 = 0; c2 < 16; ++c2) rs += hh[l * 16 + c2];
        float v = rs;
        #pragma unroll
        for (int d = 1; d < 16; d <<= 1) {
            float n = __shfl_up(v, d, 32);
            if (l >= d) v += n;
        }
        offArr[l] = v - rs;          // exclusive scan
    }
    __syncthreads();

    // D = H x U via 4 x v_wmma_f32_16x16x4_f32 (EXEC all ones: no divergence)
    const int halfsel = l >> 4;      // lane half selects K/M split
    const int Mlow    = l & 15;
    v8f c = {};
    #pragma unroll
    for (int k = 0; k < 4; ++k) {
        const int kbase = 4 * k + 2 * halfsel;
        v2f a, b;
        // A(16x4): lane l, vgpr j -> H[M = l%16][K = kbase + j]
        a.x = hh[Mlow * 16 + kbase + 0];
        a.y = hh[Mlow * 16 + kbase + 1];
        // B(4x16): lane l, vgpr j -> U[K = kbase + j][N = l%16] = (K <= N)
        b.x = (kbase + 0 <= Mlow) ? 1.0f : 0.0f;
        b.y = (kbase + 1 <= Mlow) ? 1.0f : 0.0f;
        c = __builtin_amdgcn_wmma_f32_16x16x4_f32(
                /*neg_a=*/false, a, /*neg_b=*/false, b,
                /*c_mod=*/(short)0, c, /*reuse_a=*/false, /*reuse_b=*/false);
    }

    // C/D layout: vgpr v -> row M = v + 8*(l>=16), col N = l%16
    #pragma unroll
    for (int v = 0; v < 8; ++v) {
        const int M   = v + 8 * halfsel;
        float cdf     = c[v] + offArr[M];
        float lutv    = fminf(fmaxf(roundf(cdf * (255.0f / AREA_F)), 0.0f), 255.0f);
        glut[tile * NBINS + M * 16 + Mlow] = lutv;
    }
}

// ----------- kernel 3: bilinear LUT apply + Lab->RGB (streaming) -----------
// One block per image row (2048 blocks x 256 threads, 8 px/thread). Only the
// two tile-rows y0/y1 of LUTs are needed per image row -> 16 KB LDS.
__global__ void k_apply(const float* __restrict__ x, const float* __restrict__ glut,
                        float* __restrict__ out) {
    __shared__ float llut[2 * NT * NBINS];   // 16 KB
    const int tid = threadIdx.x;
    const int h   = blockIdx.x;

    const float fyp = fminf(fmaxf((h + 0.5f) * (1.0f / TILE) - 0.5f, 0.0f), (float)(NT - 1));
    const int   y0  = (int)floorf(fyp);
    const int   y1  = min(y0 + 1, NT - 1);
    const float wy  = fyp - (float)y0;

    #pragma unroll
    for (int i = 0; i < 16; ++i) {
        const int slot = i >> 3;             // 0 -> y0 row, 1 -> y1 row
        const int tx   = i & 7;
        const int yrow = slot ? y1 : y0;
        llut[i * NBINS + tid] = glut[(yrow * NT + tx) * NBINS + tid];
    }
    __syncthreads();

    const float* __restrict__ R = x;
    const float* __restrict__ G = x + IMG_H * IMG_W;
    const float* __restrict__ B = x + 2 * IMG_H * IMG_W;
    float* __restrict__ OR = out;
    float* __restrict__ OG = out + IMG_H * IMG_W;
    float* __restrict__ OB = out + 2 * IMG_H * IMG_W;

    for (int p = 0; p < 8; ++p) {
        const int w   = p * 256 + tid;       // coalesced
        const int idx = h * IMG_W + w;

        float L, a, bb;
        rgb2lab_f(R[idx], G[idx], B[idx], L, a, bb);
        const int L8 = lab_L_to_bin(L);

        const float fxp = fminf(fmaxf((w + 0.5f) * (1.0f / TILE) - 0.5f, 0.0f), (float)(NT - 1));
        const int   x0  = (int)floorf(fxp);
        const int   x1  = min(x0 + 1, NT - 1);
        const float wx  = fxp - (float)x0;

        const float g00 = llut[x0 * NBINS + L8];
        const float g01 = llut[x1 * NBINS + L8];
        const float g10 = llut[(NT + x0) * NBINS + L8];
        const float g11 = llut[(NT + x1) * NBINS + L8];
        const float top = (1.0f - wx) * g00 + wx * g01;
        const float bot = (1.0f - wx) * g10 + wx * g11;
        const float Leq = ((1.0f - wy) * top + wy * bot) * (100.0f / 255.0f);

        // lab2rgb
        const float fy2 = (Leq + 16.0f) * (1.0f / 116.0f);
        const float fx2 = fy2 + a * (1.0f / 500.0f);
        const float fz2 = fy2 - bb * (1.0f / 200.0f);
        float xf = fx2 * fx2 * fx2; xf = (xf > LAB_EPS) ? xf : (fx2 - 16.0f / 116.0f) * (1.0f / 7.787f);
        float yf = fy2 * fy2 * fy2; yf = (yf > LAB_EPS) ? yf : (fy2 - 16.0f / 116.0f) * (1.0f / 7.787f);
        float zf = fz2 * fz2 * fz2; zf = (zf > LAB_EPS) ? zf : (fz2 - 16.0f / 116.0f) * (1.0f / 7.787f);
        const float X = xf * 0.950456f, Y = yf, Z = zf * 1.088754f;
        float ro =  3.240479f * X - 1.537150f * Y - 0.498535f * Z;
        float go = -0.969256f * X + 1.875992f * Y + 0.041556f * Z;
        float bo =  0.055648f * X - 0.204043f * Y + 1.057311f * Z;

        OR[idx] = fminf(fmaxf(ro, 0.0f), 1.0f);
        OG[idx] = fminf(fmaxf(go, 0.0f), 1.0f);
        OB[idx] = fminf(fmaxf(bo, 0.0f), 1.0f);
    }
}

// ------------------------------- launcher ----------------------------------
extern "C" void kernel_launch(void* const* d_in, const int* in_sizes, int n_in,
                              void* d_out, int out_size, void* d_ws, size_t ws_size,
                              hipStream_t stream) {
    (void)in_sizes; (void)n_in; (void)out_size; (void)ws_size;
    const float* x   = (const float*)d_in[0];
    float*       out = (float*)d_out;

    unsigned int* ghist = (unsigned int*)d_ws;                               // 64*256 u32
    float*        glut  = (float*)((char*)d_ws + 64 * NBINS * sizeof(unsigned int)); // 64*256 f32

    const int nhist = 64 * NBINS;
    k_zero <<<(nhist + 255) / 256, 256, 0, stream>>>(ghist, nhist);
    k_hist <<<1024, 256, 0, stream>>>(x, ghist);
    k_lut  <<<64,   32,  0, stream>>>(ghist, glut);
    k_apply<<<IMG_H, 256, 0, stream>>>(x, glut, out);
}